// CompGCNCov_63204738728139
// MI455X (gfx1250) — compile-verified
//
#include <hip/hip_runtime.h>

#define NN 50000      // nodes
#define NE 800000     // edges
#define DD 64         // feature dim (in == out)
#define NR 500        // relations
#define NT 1000       // time embeddings

typedef __attribute__((ext_vector_type(2))) float v2f;
typedef __attribute__((ext_vector_type(8))) float v8f;

// ---------- order-preserving float<->uint for atomicMax segment-max ----------
__device__ __forceinline__ unsigned enc_ord(float f) {
    unsigned b = __float_as_uint(f);
    return (b & 0x80000000u) ? ~b : (b | 0x80000000u);
}
__device__ __forceinline__ float dec_ord(unsigned u) {
    unsigned b = (u & 0x80000000u) ? (u & 0x7FFFFFFFu) : ~u;
    return __uint_as_float(b);
}

// ---------- init scratch: agg=0, den=0, mkey=0 (== -inf under ordering) ------
__global__ void k_init(float* __restrict__ agg, float* __restrict__ den,
                       unsigned* __restrict__ mkey) {
    int i = blockIdx.x * blockDim.x + threadIdx.x;
    if (i < NN * DD) agg[i] = 0.0f;
    if (i < NN) { den[i] = 0.0f; mkey[i] = 0u; }
}

// ---------- scalar attention projections (GEMV, L2-resident) -----------------
__global__ void k_proj(const float* __restrict__ x, const float* __restrict__ rel,
                       const float* __restrict__ temb,
                       const float* __restrict__ ah, const float* __restrict__ at,
                       const float* __restrict__ ar, const float* __restrict__ ats,
                       float* __restrict__ h_att, float* __restrict__ t_att,
                       float* __restrict__ r_att, float* __restrict__ ts_att) {
    int i = blockIdx.x * blockDim.x + threadIdx.x;
    if (i < NN) {
        const float4* xr  = (const float4*)(x + (size_t)i * DD);
        const float4* ahp = (const float4*)ah;
        const float4* atp = (const float4*)at;
        float s0 = 0.f, s1 = 0.f;
        #pragma unroll
        for (int j = 0; j < DD / 4; ++j) {
            float4 v = xr[j], a0 = ahp[j], a1 = atp[j];
            s0 = fmaf(v.x, a0.x, fmaf(v.y, a0.y, fmaf(v.z, a0.z, fmaf(v.w, a0.w, s0))));
            s1 = fmaf(v.x, a1.x, fmaf(v.y, a1.y, fmaf(v.z, a1.z, fmaf(v.w, a1.w, s1))));
        }
        h_att[i] = s0; t_att[i] = s1;
    } else if (i < NN + NR) {
        int r = i - NN;
        const float4* rr  = (const float4*)(rel + (size_t)r * DD);
        const float4* arp = (const float4*)ar;
        float s = 0.f;
        #pragma unroll
        for (int j = 0; j < DD / 4; ++j) {
            float4 v = rr[j], a0 = arp[j];
            s = fmaf(v.x, a0.x, fmaf(v.y, a0.y, fmaf(v.z, a0.z, fmaf(v.w, a0.w, s))));
        }
        r_att[r] = s;
    } else if (i < NN + NR + NT) {
        int t = i - NN - NR;
        const float4* tr  = (const float4*)(temb + (size_t)t * DD);
        const float4* atsp = (const float4*)ats;
        float s = 0.f;
        #pragma unroll
        for (int j = 0; j < DD / 4; ++j) {
            float4 v = tr[j], a0 = atsp[j];
            s = fmaf(v.x, a0.x, fmaf(v.y, a0.y, fmaf(v.z, a0.z, fmaf(v.w, a0.w, s))));
        }
        ts_att[t] = s;
    }
}

// ---------- edge logits + segment-max over dst -------------------------------
__global__ void k_edge_logit(const int* __restrict__ src, const int* __restrict__ dst,
                             const int* __restrict__ etype, const int* __restrict__ etime,
                             const float* __restrict__ h_att, const float* __restrict__ t_att,
                             const float* __restrict__ r_att, const float* __restrict__ ts_att,
                             float* __restrict__ ebuf, unsigned* __restrict__ mkey) {
    int i = blockIdx.x * blockDim.x + threadIdx.x;
    if (i >= NE) return;
    float e = h_att[src[i]] - t_att[dst[i]] + r_att[etype[i]] + ts_att[etime[i]];
    e = (e > 0.f) ? e : 0.1f * e;             // leaky_relu(0.1)
    ebuf[i] = e;
    atomicMax(&mkey[dst[i]], enc_ord(e));
}

// ---------- exp(e - max) + segment-sum denominator ---------------------------
__global__ void k_edge_exp(const int* __restrict__ dst, const unsigned* __restrict__ mkey,
                           float* __restrict__ ebuf, float* __restrict__ den) {
    int i = blockIdx.x * blockDim.x + threadIdx.x;
    if (i >= NE) return;
    int d = dst[i];
    float ex = __expf(ebuf[i] - dec_ord(mkey[d]));
    ebuf[i] = ex;
    atomicAdd(&den[d], ex);
}

// ---------- weighted edge aggregation: agg[dst] += att*(x[src]+t)*(rel+t) ----
// One wave32 per edge; lane covers dims {2*lane, 2*lane+1} -> three float2
// gathers + two f32 atomic adds. (att commutes with trans_w, so the per-edge
// GEMM collapses into this scatter-add; everything here is L2-resident.)
__global__ void k_edge_agg(const float* __restrict__ x, const float* __restrict__ rel,
                           const float* __restrict__ temb,
                           const int* __restrict__ src, const int* __restrict__ dst,
                           const int* __restrict__ etype, const int* __restrict__ etime,
                           const float* __restrict__ ebuf, const float* __restrict__ den,
                           float* __restrict__ agg) {
    long long tid = (long long)blockIdx.x * blockDim.x + threadIdx.x;
    int edge = (int)(tid >> 5);
    int lane = (int)(tid & 31);
    if (edge >= NE) return;
    int s = src[edge], d = dst[edge], r = etype[edge], t = etime[edge];
    float att = ebuf[edge] / den[d];
    const float2* xp = (const float2*)(x    + (size_t)s * DD);
    const float2* rp = (const float2*)(rel  + (size_t)r * DD);
    const float2* tp = (const float2*)(temb + (size_t)t * DD);
    float2 xv = xp[lane], rv = rp[lane], tv = tp[lane];
    float v0 = att * (xv.x + tv.x) * (rv.x + tv.x);
    float v1 = att * (xv.y + tv.y) * (rv.y + tv.y);
    float* ap = agg + (size_t)d * DD + 2 * lane;
    atomicAdd(ap,     v0);
    atomicAdd(ap + 1, v1);
}

// ---------- WMMA fp32 GEMM: out = A1@W1 (+ A2@W2), [nrows x 64] -------------
// One wave per 16x16 output tile; 4 waves/block cover the 64 output columns of
// a 16-row tile. Chains 16 (or 32) V_WMMA_F32_16X16X4_F32 ops over K=64.
// Out-of-range rows are CLAMPED (not masked): garbage A rows only affect
// garbage C rows, which the store masks — keeps the K-loop branch-free and
// EXEC all-ones for WMMA.
// A 16x4 layout: lanes 0-15 / 16-31 both hold rows 0..15; VGPR {0,1} hold
// K={0,1} (low lanes) / K={2,3} (high lanes). B 4x16 mirrors this over N.
// C/D: VGPR j -> rows j (lanes 0-15) and j+8 (lanes 16-31), lane -> column.
__global__ void k_gemm_dual(const float* __restrict__ A1, const float* __restrict__ W1,
                            const float* __restrict__ A2, const float* __restrict__ W2,
                            float* __restrict__ out, int nrows) {
    int wave = threadIdx.x >> 5;
    int lane = threadIdx.x & 31;
    int hi   = lane >> 4;           // 0: K+{0,1}, 1: K+{2,3}
    int l    = lane & 15;
    int row_base = blockIdx.x * 16;
    int row  = row_base + l;
    int rowc = (row < nrows) ? row : (nrows - 1);   // clamp, no predication
    int n    = wave * 16 + l;

    // per-lane A pointer: contiguous 8B pair {kk, kk+1} each iteration
    const v2f* a1p = (const v2f*)(A1 + (size_t)rowc * DD + 2 * hi);

    v8f c = {};
    #pragma unroll
    for (int k0 = 0; k0 < DD; k0 += 4) {
        int kk = k0 + 2 * hi;
        v2f a = a1p[k0 >> 1];               // A[rowc][kk..kk+1], one b64 load
        v2f b;
        b.x = W1[kk * DD + n];              // coalesced across lanes
        b.y = W1[(kk + 1) * DD + n];
        c = __builtin_amdgcn_wmma_f32_16x16x4_f32(false, a, false, b,
                                                  (short)0, c, false, false);
    }
    if (A2) {   // uniform (scalar) branch: EXEC stays all-ones for WMMA
        const v2f* a2p = (const v2f*)(A2 + (size_t)rowc * DD + 2 * hi);
        #pragma unroll
        for (int k0 = 0; k0 < DD; k0 += 4) {
            int kk = k0 + 2 * hi;
            v2f a = a2p[k0 >> 1];
            v2f b;
            b.x = W2[kk * DD + n];
            b.y = W2[(kk + 1) * DD + n];
            c = __builtin_amdgcn_wmma_f32_16x16x4_f32(false, a, false, b,
                                                      (short)0, c, false, false);
        }
    }
    #pragma unroll
    for (int j = 0; j < 8; ++j) {
        int r = row_base + j + 8 * hi;
        if (r < nrows) out[(size_t)r * DD + n] = c[j];
    }
}

extern "C" void kernel_launch(void* const* d_in, const int* in_sizes, int n_in,
                              void* d_out, int out_size, void* d_ws, size_t ws_size,
                              hipStream_t stream) {
    const float* x       = (const float*)d_in[0];
    const float* rel     = (const float*)d_in[1];
    const float* temb    = (const float*)d_in[2];
    const int*   src     = (const int*)d_in[3];
    const int*   dst     = (const int*)d_in[4];
    const int*   etype   = (const int*)d_in[5];
    const int*   etime   = (const int*)d_in[6];
    const float* trans_w = (const float*)d_in[7];
    const float* loop_w  = (const float*)d_in[8];
    const float* w_rel   = (const float*)d_in[9];
    const float* ah      = (const float*)d_in[10];
    const float* at      = (const float*)d_in[11];
    const float* ar      = (const float*)d_in[12];
    const float* ats     = (const float*)d_in[13];
    float* out = (float*)d_out;

    // workspace layout (all 4-byte elements, ~17 MB total)
    float*    h_att  = (float*)d_ws;              // NN
    float*    t_att  = h_att + NN;                // NN
    float*    r_att  = t_att + NN;                // NR
    float*    ts_att = r_att + NR;                // NT
    unsigned* mkey   = (unsigned*)(ts_att + NT);  // NN
    float*    den    = (float*)(mkey + NN);       // NN
    float*    ebuf   = den + NN;                  // NE
    float*    agg    = ebuf + NE;                 // NN*DD

    k_init<<<(NN * DD + 255) / 256, 256, 0, stream>>>(agg, den, mkey);
    k_proj<<<(NN + NR + NT + 255) / 256, 256, 0, stream>>>(
        x, rel, temb, ah, at, ar, ats, h_att, t_att, r_att, ts_att);
    k_edge_logit<<<(NE + 255) / 256, 256, 0, stream>>>(
        src, dst, etype, etime, h_att, t_att, r_att, ts_att, ebuf, mkey);
    k_edge_exp<<<(NE + 255) / 256, 256, 0, stream>>>(dst, mkey, ebuf, den);
    {
        long long total = (long long)NE * 32;
        int blocks = (int)((total + 255) / 256);
        k_edge_agg<<<blocks, 256, 0, stream>>>(x, rel, temb, src, dst, etype, etime,
                                               ebuf, den, agg);
    }
    // x_out = agg @ trans_w + x @ loop_w
    k_gemm_dual<<<(NN + 15) / 16, 128, 0, stream>>>(agg, trans_w, x, loop_w, out, NN);
    // rel_out = rel_repr @ w_rel
    k_gemm_dual<<<(NR + 15) / 16, 128, 0, stream>>>(rel, w_rel, nullptr, nullptr,
                                                    out + (size_t)NN * DD, NR);
}